// SuperGaussianPSF_62835371541137
// MI455X (gfx1250) — compile-verified
//
#include <hip/hip_runtime.h>
#include <cstdint>
#include <cstddef>

// ---------------------------------------------------------------------------
// SuperGaussianPSF: per-batch 5x5 generalized-Gaussian depthwise conv,
// reflect padding, fp32, fixed shape (B=16, C=4, H=W=1024).
// HBM-bound (512 MB traffic, ~22us floor @ 23.3 TB/s).
// Strategy: TDM (tensor_load_to_lds) DMA of 68x68 tiles into LDS for interior
// tiles, VALU reflect-fill for border tiles, 4x4 register-blocked FMA compute.
// ---------------------------------------------------------------------------

#define RADIUS   2
#define TILE     64
#define LW       (TILE + 2 * RADIUS)   // 68
#define NTHREADS 256

// Fixed problem geometry (from the reference setup_inputs).
#define DIM_H    1024
#define DIM_W    1024
#define DIM_C    4
#define TPR      (DIM_W / TILE)        // tiles per row = 16
#define TPC      (DIM_H / TILE)        // tiles per col = 16

// Toolchain discrimination: amdgpu-toolchain (clang-23) ships the TDM header
// and uses the 6-arg builtin; ROCm 7.2 (clang-22) uses the 5-arg builtin.
#if defined(__has_include)
#  if __has_include(<hip/amd_detail/amd_gfx1250_TDM.h>)
#    define TDM_6ARG 1
#  else
#    define TDM_6ARG 0
#  endif
#else
#  define TDM_6ARG 0
#endif

#if defined(__has_builtin)
#  if __has_builtin(__builtin_amdgcn_tensor_load_to_lds)
#    define HAVE_TDM 1
#  else
#    define HAVE_TDM 0
#  endif
#else
#  define HAVE_TDM 0
#endif

// Device-pass-only diagnostics so the compile log proves which path lowered.
#if defined(__AMDGCN__)
#  if HAVE_TDM && TDM_6ARG
#    warning "gfx1250 device pass: TDM path ENABLED (6-arg builtin)"
#  elif HAVE_TDM
#    warning "gfx1250 device pass: TDM path ENABLED (5-arg builtin)"
#  else
#    warning "gfx1250 device pass: TDM path DISABLED (builtin missing)"
#  endif
#endif

typedef __attribute__((ext_vector_type(4))) unsigned int uint32x4;
typedef __attribute__((ext_vector_type(8))) int          int32x8;
typedef __attribute__((ext_vector_type(4))) int          int32x4;

// ---------------------------------------------------------------------------
// Main kernel (first in .text for the disasm snippet): tiled 5x5 conv.
// One block = one 64x64 output tile of one plane. 1D grid:
//   blockIdx.x = plane * 256 + by * 16 + bx
// ---------------------------------------------------------------------------
__global__ __launch_bounds__(NTHREADS) void psf_conv5x5(
    const float* __restrict__ x, const float* __restrict__ kw,
    float* __restrict__ out) {
  __shared__ float tile[LW * LW];

  const unsigned bid = blockIdx.x;
  const int bx    = (int)(bid & (TPR - 1));
  const int by    = (int)((bid >> 4) & (TPC - 1));
  const int plane = (int)(bid >> 8);
  const int gx0   = bx << 6;
  const int gy0   = by << 6;
  const float* __restrict__ src = x + ((size_t)plane << 20);  // plane*H*W

  // ---- Stage input tile (68x68) into LDS: TDM DMA fast path first so the
  // ---- DMA is in flight while we fetch the 25 weights into SGPRs.
  bool tdm_done = false;
#if HAVE_TDM
  const bool interior =
      ((unsigned)(bx - 1) < (unsigned)(TPR - 2)) &&
      ((unsigned)(by - 1) < (unsigned)(TPC - 2));
  if (interior) {
    tdm_done = true;
    if ((threadIdx.x >> 5) == 0) {  // wave 0 of the block issues one TDM DMA
      const unsigned lds_off = (unsigned)(uintptr_t)(void*)&tile[0];
      const unsigned long long ga = (unsigned long long)(uintptr_t)(
          src + (size_t)(gy0 - RADIUS) * DIM_W + (gx0 - RADIUS));
      // D# group0: count=1 | lds_addr | global_addr[56:0] | type=2
      uint32x4 g0 = {1u, lds_off, (unsigned)(ga & 0xFFFFFFFFu),
                     (unsigned)((ga >> 32) & 0x01FFFFFFu) | (2u << 30)};
      // D# group1: data_size=4B; tensor_dim0/1 = LW (tile fully in-bounds by
      // construction); tile_dim0/1 = LW; tile_dim2=0 (2D); dim0_stride = W.
      int32x8 g1 = {(int)(2u << 16),
                    (int)((unsigned)LW << 16),    // tensor_dim0[15:0]
                    (int)((unsigned)LW << 16),    // dim0 hi=0 | tensor_dim1 lo
                    (int)((unsigned)LW << 16),    // dim1 hi=0 | tile_dim0
                    (int)(unsigned)LW,            // tile_dim1 | tile_dim2=0
                    (int)(unsigned)DIM_W,         // tensor_dim0_stride[31:0]
                    0, 0};
      int32x4 z4 = {0, 0, 0, 0};
#if TDM_6ARG
      int32x8 z8 = {0, 0, 0, 0, 0, 0, 0, 0};
      __builtin_amdgcn_tensor_load_to_lds(g0, g1, z4, z4, z8, 0);
#else
      __builtin_amdgcn_tensor_load_to_lds(g0, g1, z4, z4, 0);
#endif
      __builtin_amdgcn_s_wait_tensorcnt(0);
    }
  }
#endif

  // Per-batch 25 weights: block-uniform address -> scalar loads into SGPRs.
  const int b = plane >> 2;  // C = 4 channels per batch
  float w[25];
#pragma unroll
  for (int k = 0; k < 25; ++k) w[k] = kw[b * 25 + k];

  if (!tdm_done) {
    // Border tiles (or no-TDM toolchain): cooperative reflect-padded fill.
    for (int idx = threadIdx.x; idx < LW * LW; idx += NTHREADS) {
      const int ly = idx / LW;
      const int lx = idx - ly * LW;
      int gy = gy0 + ly - RADIUS;
      int gx = gx0 + lx - RADIUS;
      gy = (gy < 0) ? -gy : ((gy >= DIM_H) ? 2 * DIM_H - 2 - gy : gy);
      gx = (gx < 0) ? -gx : ((gx >= DIM_W) ? 2 * DIM_W - 2 - gx : gx);
      tile[idx] = src[(size_t)gy * DIM_W + gx];
    }
  }
  __syncthreads();

  // ---- Compute: each thread does a 4x4 output block (register blocking) ----
  const int tx = (threadIdx.x & 15) << 2;  // 0..60, float4-aligned
  const int ty = (threadIdx.x >> 4) << 2;  // 0..60

  float acc[16];
#pragma unroll
  for (int k = 0; k < 16; ++k) acc[k] = 0.0f;

#pragma unroll
  for (int row = 0; row < 8; ++row) {     // patch rows ty+0 .. ty+7
    const float* p = &tile[(ty + row) * LW + tx];
    const float4 lo = *(const float4*)(p);
    const float4 hi = *(const float4*)(p + 4);
    const float f[8] = {lo.x, lo.y, lo.z, lo.w, hi.x, hi.y, hi.z, hi.w};
#pragma unroll
    for (int r = 0; r < 4; ++r) {         // output row within 4x4 block
      const int ki = row - r;             // kernel row index
      if (ki >= 0 && ki < 5) {
#pragma unroll
        for (int c = 0; c < 4; ++c) {     // output col within 4x4 block
          float s = f[c] * w[ki * 5 + 0];
          s = fmaf(f[c + 1], w[ki * 5 + 1], s);
          s = fmaf(f[c + 2], w[ki * 5 + 2], s);
          s = fmaf(f[c + 3], w[ki * 5 + 3], s);
          s = fmaf(f[c + 4], w[ki * 5 + 4], s);
          acc[r * 4 + c] += s;
        }
      }
    }
  }

  float* __restrict__ dst = out + ((size_t)plane << 20) +
                            (size_t)(gy0 + ty) * DIM_W + (gx0 + tx);
#pragma unroll
  for (int r = 0; r < 4; ++r) {
    float4 o = {acc[r * 4 + 0], acc[r * 4 + 1], acc[r * 4 + 2], acc[r * 4 + 3]};
    *(float4*)(dst + (size_t)r * DIM_W) = o;
  }
}

// ---------------------------------------------------------------------------
// Kernel 2: build the 16 normalized 5x5 kernels (one block per batch).
// kernel[i][j] = exp(-(|lin[i]*g/wx|^kx + |lin[j]*g/wy|^ky)), lin = [-2..2]
// ---------------------------------------------------------------------------
__global__ void psf_make_kernels(const float* __restrict__ grid_size,
                                 const float* __restrict__ log_wx,
                                 const float* __restrict__ log_wy,
                                 const float* __restrict__ log_kx,
                                 const float* __restrict__ log_ky,
                                 float* __restrict__ kw) {
  __shared__ float vals[32];
  const int b = blockIdx.x;
  const int t = threadIdx.x;

  const float g  = grid_size[b];
  const float wx = expf(log_wx[0]) + 0.001f * g;
  const float wy = expf(log_wy[0]) + 0.001f * g;
  const float kx = expf(log_kx[0]) + 1.0f;
  const float ky = expf(log_ky[0]) + 1.0f;

  float v = 0.0f;
  if (t < 25) {
    const int i = t / 5;           // kernel H index (xx axis, meshgrid 'ij')
    const int j = t - i * 5;       // kernel W index (yy axis)
    const float sx = (float)(i - RADIUS) * g;
    const float sy = (float)(j - RADIUS) * g;
    v = expf(-(powf(fabsf(sx / wx), kx) + powf(fabsf(sy / wy), ky)));
  }
  vals[t] = v;
  __syncthreads();
  if (t == 0) {
    float s = 0.0f;
    for (int k = 0; k < 25; ++k) s += vals[k];
    vals[31] = s;
  }
  __syncthreads();
  if (t < 25) kw[b * 25 + t] = vals[t] / vals[31];
}

// ---------------------------------------------------------------------------
extern "C" void kernel_launch(void* const* d_in, const int* in_sizes, int n_in,
                              void* d_out, int out_size, void* d_ws,
                              size_t ws_size, hipStream_t stream) {
  const float* x         = (const float*)d_in[0];
  const float* grid_size = (const float*)d_in[1];
  const float* log_wx    = (const float*)d_in[2];
  const float* log_wy    = (const float*)d_in[3];
  const float* log_kx    = (const float*)d_in[4];
  const float* log_ky    = (const float*)d_in[5];
  float* out = (float*)d_out;
  float* kw  = (float*)d_ws;  // B * 25 floats of scratch

  const int B = in_sizes[1];                       // 16
  const int C = in_sizes[0] / (B * DIM_H * DIM_W); // 4

  psf_make_kernels<<<B, 32, 0, stream>>>(grid_size, log_wx, log_wy, log_kx,
                                         log_ky, kw);

  const int nblocks = TPR * TPC * B * C;  // 16*16*64 = 16384
  psf_conv5x5<<<nblocks, NTHREADS, 0, stream>>>(x, kw, out);
}